// LightGCN_47931835023877
// MI455X (gfx1250) — compile-verified
//
#include <hip/hip_runtime.h>
#include <cstdint>
#include <cstddef>

#define NUM_USERS 100000
#define NUM_ITEMS 50000
#define N_NODES   (NUM_USERS + NUM_ITEMS)
#define EMB       64
#define N_LAYERS  3

#define WPB 8            // waves per block (wave32)
#define TPB (WPB * 32)   // 256 threads
#define CH  128          // edges staged per wave per chunk

// ---- CDNA5 async global->LDS (ASYNCcnt), GV addressing, 16B per lane ----
__device__ __forceinline__ void async_g2lds_b128(uint32_t lds_byte_addr, uint64_t gaddr) {
    asm volatile("global_load_async_to_lds_b128 %0, %1, off"
                 :: "v"(lds_byte_addr), "v"(gaddr)
                 : "memory");
}
__device__ __forceinline__ void wait_async_le3() {   // oldest 3 of <=6 done
    asm volatile("s_wait_asynccnt 0x3" ::: "memory");
}
__device__ __forceinline__ void wait_async_0() {
    asm volatile("s_wait_asynccnt 0x0" ::: "memory");
}
// ---- guaranteed no-return f32 atomic add (STOREcnt path, no CAS loop) ----
__device__ __forceinline__ void atomic_add_f32_nr(float* p, float v) {
    asm volatile("global_atomic_add_f32 %0, %1, off"
                 :: "v"((uint64_t)(size_t)p), "v"(v)
                 : "memory");
}

// ---- COO SpMM: nxt[dst] += val * cur[src] ----
// A wave stages CH=128 edge triples into LDS with 3 async B128 loads
// (double-buffered: next chunk streams in while this chunk's atomics issue).
// Then the 32 lanes process one edge at a time: lane k owns feature elements
// [2k,2k+1], so each 64-float row is one coalesced 256B gather plus two
// fire-and-forget global_atomic_add_f32. Both embedding buffers are
// L2-resident on MI455X (2 x 38.4 MB << 192 MB).
__global__ void __launch_bounds__(TPB)
lgcn_spmm_kernel(const int* __restrict__ src, const int* __restrict__ dst,
                 const float* __restrict__ val, const float* __restrict__ cur,
                 float* __restrict__ nxt, int nEdges) {
    __shared__ int   s_src[2][WPB][CH];
    __shared__ int   s_dst[2][WPB][CH];
    __shared__ float s_val[2][WPB][CH];

    const int lane   = threadIdx.x & 31;
    const int wave   = threadIdx.x >> 5;
    const int waveId = blockIdx.x * WPB + wave;
    const int stride = gridDim.x * WPB * CH;
    const int nE4    = nEdges & ~3;           // granule-aligned main region

    // stage a chunk: lane i moves the 4-edge granule [gbase+4i, gbase+4i+3]
    auto stage = [&](int gbase, int buf) {
        int g = gbase + 4 * lane;
        int gmax = nE4 - 4;
        if (g > gmax) g = gmax;               // clamp: dupes land at j>=cnt
        async_g2lds_b128((uint32_t)(size_t)&s_src[buf][wave][4 * lane],
                         (uint64_t)(size_t)(src + g));
        async_g2lds_b128((uint32_t)(size_t)&s_dst[buf][wave][4 * lane],
                         (uint64_t)(size_t)(dst + g));
        async_g2lds_b128((uint32_t)(size_t)&s_val[buf][wave][4 * lane],
                         (uint64_t)(size_t)(val + g));
    };

    int base = waveId * CH;
    int buf  = 0;
    if (base < nE4) stage(base, buf);

    for (; base < nE4; base += stride, buf ^= 1) {
        const int nextBase = base + stride;
        if (nextBase < nE4) {
            stage(nextBase, buf ^ 1);
            wait_async_le3();                 // in-order: current 3 complete
        } else {
            wait_async_0();
        }

        int rem = nE4 - base;
        int cnt = rem < CH ? rem : CH;

        // warm caches with a sample of this chunk's source rows
        {
            int pj = 4 * lane;
            if (pj > cnt - 1) pj = cnt - 1;
            __builtin_prefetch(cur + (size_t)s_src[buf][wave][pj] * EMB, 0, 1);
        }

        for (int j = 0; j < cnt; ++j) {
            const int   sj = s_src[buf][wave][j];   // LDS broadcast reads
            const int   dj = s_dst[buf][wave][j];
            const float vj = s_val[buf][wave][j];
            const float2 x = *(const float2*)(cur + (size_t)sj * EMB + lane * 2);
            float* out = nxt + (size_t)dj * EMB + lane * 2;
            atomic_add_f32_nr(out + 0, vj * x.x);
            atomic_add_f32_nr(out + 1, vj * x.y);
        }
    }

    // remainder edges (nEdges % 4, zero for this problem size): scalar path
    if (blockIdx.x == 0 && wave == 0) {
        for (int e = nE4; e < nEdges; ++e) {
            const int   sj = src[e];
            const int   dj = dst[e];
            const float vj = val[e];
            const float2 x = *(const float2*)(cur + (size_t)sj * EMB + lane * 2);
            float* out = nxt + (size_t)dj * EMB + lane * 2;
            atomic_add_f32_nr(out + 0, vj * x.x);
            atomic_add_f32_nr(out + 1, vj * x.y);
        }
    }
}

// ---- concat(user_emb, item_emb) -> cur, and initialize accumulator ----
__global__ void __launch_bounds__(256)
lgcn_init_kernel(const float* __restrict__ ue, const float* __restrict__ ie,
                 float* __restrict__ cur, float* __restrict__ acc,
                 int n4user, int n4total) {
    int i = blockIdx.x * blockDim.x + threadIdx.x;
    if (i >= n4total) return;
    float4 v = (i < n4user) ? ((const float4*)ue)[i]
                            : ((const float4*)ie)[i - n4user];
    ((float4*)cur)[i] = v;
    ((float4*)acc)[i] = v;
}

__global__ void __launch_bounds__(256)
lgcn_zero_kernel(float* __restrict__ p, int n4) {
    int i = blockIdx.x * blockDim.x + threadIdx.x;
    if (i >= n4) return;
    ((float4*)p)[i] = make_float4(0.f, 0.f, 0.f, 0.f);
}

// acc = (acc + nxt) * scale   (scale = 1 inner layers, 1/4 fused at the end)
__global__ void __launch_bounds__(256)
lgcn_accscale_kernel(float* __restrict__ acc, const float* __restrict__ nxt,
                     float scale, int n4) {
    int i = blockIdx.x * blockDim.x + threadIdx.x;
    if (i >= n4) return;
    float4 a = ((float4*)acc)[i];
    float4 b = ((const float4*)nxt)[i];
    a.x = (a.x + b.x) * scale;
    a.y = (a.y + b.y) * scale;
    a.z = (a.z + b.z) * scale;
    a.w = (a.w + b.w) * scale;
    ((float4*)acc)[i] = a;
}

extern "C" void kernel_launch(void* const* d_in, const int* in_sizes, int n_in,
                              void* d_out, int out_size, void* d_ws, size_t ws_size,
                              hipStream_t stream) {
    // inputs: 0 user_indices (unused), 1 item_indices (unused),
    //         2 adj_src, 3 adj_dst, 4 adj_vals, 5 user_emb, 6 item_emb
    const int*   adj_src  = (const int*)d_in[2];
    const int*   adj_dst  = (const int*)d_in[3];
    const float* adj_val  = (const float*)d_in[4];
    const float* user_emb = (const float*)d_in[5];
    const float* item_emb = (const float*)d_in[6];
    const int    nEdges   = in_sizes[2];

    const size_t nodeFloats = (size_t)N_NODES * EMB;      // 9.6M floats
    float* bufA = (float*)d_ws;                           // 38.4 MB
    float* bufB = bufA + nodeFloats;                      // 38.4 MB
    float* acc  = (float*)d_out;

    const int n4total = (int)(nodeFloats / 4);            // 2,400,000
    const int n4user  = NUM_USERS * EMB / 4;              // 1,600,000
    const int eltBlocks = (n4total + 255) / 256;

    lgcn_init_kernel<<<eltBlocks, 256, 0, stream>>>(user_emb, item_emb,
                                                    bufA, acc, n4user, n4total);

    float* cur = bufA;
    float* nxt = bufB;
    for (int layer = 0; layer < N_LAYERS; ++layer) {
        lgcn_zero_kernel<<<eltBlocks, 256, 0, stream>>>(nxt, n4total);
        lgcn_spmm_kernel<<<4096, TPB, 0, stream>>>(adj_src, adj_dst, adj_val,
                                                   cur, nxt, nEdges);
        const float scale = (layer == N_LAYERS - 1)
                          ? (1.0f / (float)(N_LAYERS + 1)) : 1.0f;
        lgcn_accscale_kernel<<<eltBlocks, 256, 0, stream>>>(acc, nxt, scale,
                                                            n4total);
        float* t = cur; cur = nxt; nxt = t;
    }
}